// DynamicScatterRouter_19739669692887
// MI455X (gfx1250) — compile-verified
//
#include <hip/hip_runtime.h>
#include <hip/hip_bf16.h>

#define NTOK   8192
#define DMODEL 1024
#define NEXP   8
#define WAVE   32   // gfx1250 is wave32

typedef int   v4i __attribute__((ext_vector_type(4)));
typedef float v4f __attribute__((ext_vector_type(4)));
typedef __attribute__((address_space(1))) v4i gv4i_t;   // global int4
typedef __attribute__((address_space(3))) v4i lv4i_t;   // LDS int4

// ---- CDNA5 async global<->LDS path (guarded; fallback = plain b128 copies) ----
#if defined(__has_builtin)
#if __has_builtin(__builtin_amdgcn_global_load_async_to_lds_b128) && \
    __has_builtin(__builtin_amdgcn_global_store_async_from_lds_b128)
#define HAVE_ASYNC_LDS 1
#endif
#endif
#ifndef HAVE_ASYNC_LDS
#define HAVE_ASYNC_LDS 0
#endif

#if defined(__has_builtin)
#if __has_builtin(__builtin_amdgcn_s_wait_asynccnt)
#define WAIT_ASYNCCNT0() __builtin_amdgcn_s_wait_asynccnt(0)
#endif
#endif
#ifndef WAIT_ASYNCCNT0
#define WAIT_ASYNCCNT0() asm volatile("s_wait_asynccnt 0" ::: "memory")
#endif

// CPol temporal hint: TH bits [2:0]; 1 = NT (non-temporal)
#define CPOL_RT 0
#define CPOL_NT 1

// -----------------------------------------------------------------------------
// Kernel: scatter (defined FIRST so the disasm snippet shows the async ops).
// One block per token; stage the 4KB row in LDS via async load, then fan out
// async NT stores from LDS to each selected expert's destination row.
// -----------------------------------------------------------------------------
__global__ __launch_bounds__(256) void scatter_kernel(const float* __restrict__ in_flow,
                                                      const int* __restrict__ ws_loc,
                                                      float* __restrict__ out) {
    __shared__ float s_row[DMODEL];    // 4 KB
    __shared__ int   s_loc[NEXP];

    const int tok = blockIdx.x;
    const int tid = threadIdx.x;

    const float* src = in_flow + (size_t)tok * DMODEL + (size_t)tid * 4;

    if (tid < NEXP) s_loc[tid] = ws_loc[(size_t)tok * NEXP + tid];

#if HAVE_ASYNC_LDS
    __builtin_amdgcn_global_load_async_to_lds_b128(
        (gv4i_t*)src, (lv4i_t*)&s_row[tid * 4], 0, CPOL_RT);
    WAIT_ASYNCCNT0();                  // row resident in LDS (per-wave)
    __syncthreads();                   // s_loc visible to all waves

    #pragma unroll
    for (int e = 0; e < NEXP; ++e) {
        int lp = s_loc[e];
        if (lp > 0) {
            float* dst = out + ((size_t)e * NTOK + (size_t)(lp - 1)) * DMODEL
                             + (size_t)tid * 4;
            __builtin_amdgcn_global_store_async_from_lds_b128(
                (gv4i_t*)dst, (lv4i_t*)&s_row[tid * 4], 0, CPOL_NT);
        }
    }
    WAIT_ASYNCCNT0();                  // drain before LDS/kernel teardown
#else
    // Fallback: register-staged b128 copy (no LDS needed for the data path)
    v4f v = *(const v4f*)src;
    __syncthreads();                   // s_loc visible
    #pragma unroll
    for (int e = 0; e < NEXP; ++e) {
        int lp = s_loc[e];
        if (lp > 0) {
            float* dst = out + ((size_t)e * NTOK + (size_t)(lp - 1)) * DMODEL
                             + (size_t)tid * 4;
            __builtin_nontemporal_store(v, (v4f*)dst);
        }
    }
    (void)s_row;
#endif
}

// -----------------------------------------------------------------------------
// Kernel: mask + per-expert cumsum (loc) + loads.
// Single workgroup of 1024 threads (32 waves), 8 iterations over 8192 tokens.
// ws_loc[tok*8+e] = loc+1 if selected else 0.  loads_out[e] = column sums.
// -----------------------------------------------------------------------------
__global__ __launch_bounds__(1024) void scan_kernel(const float* __restrict__ gates,
                                                    int* __restrict__ ws_loc,
                                                    int* __restrict__ loads_out) {
    __shared__ int s_wave[32][NEXP];  // per-wave totals -> exclusive offsets (in place)
    __shared__ int s_carry[NEXP];     // running global offset per expert

    const int tid  = threadIdx.x;
    const int wave = tid >> 5;
    const int lane = tid & (WAVE - 1);
    const unsigned lanemask_lt = (1u << lane) - 1u;

    if (tid < NEXP) s_carry[tid] = 0;
    __syncthreads();

    for (int base = 0; base < NTOK; base += 1024) {
        const int tok = base + tid;

        // mask bits for this token (bit e = gates[tok][e] > 0); residual -> expert 0
        unsigned mbits = 0u;
        #pragma unroll
        for (int e = 0; e < NEXP; ++e) {
            float g = gates[(size_t)tok * NEXP + e];
            mbits |= (g > 0.0f) ? (1u << e) : 0u;
        }
        if (mbits == 0u) mbits = 1u;

        // per-expert wave32 ballots: within-wave exclusive prefix + wave totals
        int myoff[NEXP];
        int wtot[NEXP];
        #pragma unroll
        for (int e = 0; e < NEXP; ++e) {
            unsigned b = (unsigned)__ballot((mbits >> e) & 1u);
            myoff[e] = __popc(b & lanemask_lt);
            wtot[e]  = __popc(b);
        }
        if (lane == 0) {
            #pragma unroll
            for (int e = 0; e < NEXP; ++e) s_wave[wave][e] = wtot[e];
        }
        __syncthreads();

        // threads 0..7 (one per expert) scan the 32 wave totals serially
        if (tid < NEXP) {
            int run = s_carry[tid];
            #pragma unroll
            for (int w = 0; w < 32; ++w) {
                int t = s_wave[w][tid];
                s_wave[w][tid] = run;   // now holds exclusive wave offset
                run += t;
            }
            s_carry[tid] = run;
        }
        __syncthreads();

        #pragma unroll
        for (int e = 0; e < NEXP; ++e) {
            int loc = s_wave[wave][e] + myoff[e];
            ws_loc[(size_t)tok * NEXP + e] = ((mbits >> e) & 1u) ? (loc + 1) : 0;
        }
        __syncthreads();  // protect s_wave before next iteration overwrites it
    }

    if (tid < NEXP) loads_out[tid] = s_carry[tid];
}

// -----------------------------------------------------------------------------
// Kernel: zero exactly the rows r >= loads[e] (the scatter fills r < loads[e]).
// One block per (expert,row); 256 threads x 16B NT store = 4KB row.
// -----------------------------------------------------------------------------
__global__ __launch_bounds__(256) void zero_tail_kernel(float* __restrict__ out,
                                                        const int* __restrict__ loads) {
    const int blk = blockIdx.x;        // 0 .. NEXP*NTOK-1
    const int e   = blk >> 13;         // / 8192
    const int r   = blk & (NTOK - 1);
    if (r < loads[e]) return;          // scatter covers this row
    v4f z = (v4f)(0.0f);
    float* dst = out + ((size_t)e * NTOK + (size_t)r) * DMODEL + (size_t)threadIdx.x * 4;
    __builtin_nontemporal_store(z, (v4f*)dst);
}

// -----------------------------------------------------------------------------
extern "C" void kernel_launch(void* const* d_in, const int* in_sizes, int n_in,
                              void* d_out, int out_size, void* d_ws, size_t ws_size,
                              hipStream_t stream) {
    const float* in_flow = (const float*)d_in[0];   // (8192, 1024) f32
    const float* gates   = (const float*)d_in[1];   // (8192, 8)    f32

    float* out_flows = (float*)d_out;                                    // (8,8192,1024) f32
    int*   loads_out = (int*)((float*)d_out + (size_t)NEXP * NTOK * DMODEL); // (8,) i32 tail

    int* ws_loc = (int*)d_ws;   // 8192*8 ints = 256 KB

    scan_kernel<<<1, 1024, 0, stream>>>(gates, ws_loc, loads_out);
    zero_tail_kernel<<<NEXP * NTOK, 256, 0, stream>>>(out_flows, loads_out);
    scatter_kernel<<<NTOK, 256, 0, stream>>>(in_flow, ws_loc, out_flows);
}